// ALLonBert_v2_1322849927775
// MI455X (gfx1250) — compile-verified
//
#include <hip/hip_runtime.h>
#include <hip/hip_bf16.h>

// Problem constants (match reference)
#define BB    16
#define SS    4096
#define HH    1024
#define NSEG  64
#define SEP_ID 102

typedef __attribute__((ext_vector_type(2))) float v2f;
typedef __attribute__((ext_vector_type(4))) float v4f;
typedef __attribute__((ext_vector_type(8))) float v8f;

// ---------------------------------------------------------------------------
// Kernel A: per-row SEP scan -> per-segment [start, end) token ranges.
// One wave (32 lanes) per batch row. Ballot-based ordered position extraction.
// Segment 0 = [1, sep0); segment k = (sep_{k-1}, sep_k); seg >= nsep -> empty.
// Reproduces the reference 'valid' mask (no SEP, pos>=1, seg < num_seps)
// as contiguous ranges.
// ---------------------------------------------------------------------------
__global__ __launch_bounds__(32) void seg_scan_kernel(
    const int* __restrict__ ids,
    int* __restrict__ seg_start,
    int* __restrict__ seg_end) {
  const int b    = blockIdx.x;
  const int lane = threadIdx.x;
  __shared__ int sep_pos[NSEG];

  const int* row = ids + (size_t)b * SS;
  int nsep = 0;
  for (int base = 0; base < SS; base += 32) {
    const int  pos    = base + lane;
    const bool is_sep = (row[pos] == SEP_ID);
    const unsigned mask = (unsigned)__ballot(is_sep);   // wave32: low 32 bits
    if (is_sep) {
      const int idx = nsep + __popc(mask & ((1u << lane) - 1u));
      if (idx < NSEG) sep_pos[idx] = pos;
    }
    nsep += __popc(mask);
  }
  __syncthreads();

  for (int seg = lane; seg < NSEG; seg += 32) {
    int st = 1, en = 1;                    // empty range by default
    if (seg < nsep) {
      en = sep_pos[seg];
      st = (seg == 0) ? 1 : (sep_pos[seg - 1] + 1);
      if (st < 1) st = 1;
      if (en < st) en = st;
    }
    const int fs = b * NSEG + seg;
    seg_start[fs] = st;
    seg_end[fs]   = en;
  }
}

// ---------------------------------------------------------------------------
// Kernel B: the 256 MB streaming pass. One block per flat segment (1024
// blocks x 8 waves). Thread t owns floats [4t, 4t+4) of H -> coalesced
// 512B/wave b128 loads, non-temporal (data is consumed exactly once).
// 2x unrolled with independent accumulators for memory-level parallelism.
// ---------------------------------------------------------------------------
__global__ __launch_bounds__(256) void seg_mean_kernel(
    const float* __restrict__ hidden,
    const int* __restrict__ seg_start,
    const int* __restrict__ seg_end,
    float* __restrict__ means) {
  const int fs = blockIdx.x;        // flat segment: b*NSEG + seg
  const int b  = fs >> 6;           // / NSEG
  const int t  = threadIdx.x;       // 0..255, covers H via float4
  const int st = seg_start[fs];
  const int en = seg_end[fs];

  const float* __restrict__ base =
      hidden + (size_t)b * SS * HH + (size_t)t * 4;

  v4f acc0 = {0.f, 0.f, 0.f, 0.f};
  v4f acc1 = {0.f, 0.f, 0.f, 0.f};
  int s = st;
  for (; s + 1 < en; s += 2) {
    const v4f v0 =
        __builtin_nontemporal_load((const v4f*)(base + (size_t)s * HH));
    const v4f v1 =
        __builtin_nontemporal_load((const v4f*)(base + (size_t)(s + 1) * HH));
    acc0 += v0;
    acc1 += v1;
  }
  if (s < en) {
    acc0 += __builtin_nontemporal_load((const v4f*)(base + (size_t)s * HH));
  }
  v4f acc = acc0 + acc1;

  const int   cnt = en - st;
  const float inv = 1.0f / (float)(cnt > 0 ? cnt : 1);
  acc *= inv;
  *(v4f*)(means + (size_t)fs * HH + (size_t)t * 4) = acc;
}

// ---------------------------------------------------------------------------
// Kernel C: classifier GEMM logits = means @ W^T + bias via
// V_WMMA_F32_16X16X4_F32. One wave per 16-row M-tile (64 tiles), 256 K-steps.
// A (16x4 f32): lane&15 = M row; lane>>4 selects K {0,1} vs {2,3} halves.
// B (4x16 f32): lane&15 = N col (only N<2 hold W rows, rest zeroed via
// branchless select -> EXEC stays all-1s through the loop, as WMMA requires).
// C/D (16x16 f32, 8 VGPRs): row = r + 8*(lane>=16), col = lane&15.
// ---------------------------------------------------------------------------
__global__ __launch_bounds__(32) void wmma_classifier_kernel(
    const float* __restrict__ means,
    const float* __restrict__ W,       // [2, H]
    const float* __restrict__ bias,    // [2]
    float* __restrict__ out) {         // [B*NSEG, 2]
  const int tile = blockIdx.x;         // 0..(B*NSEG/16 - 1)
  const int lane = threadIdx.x;        // wave32
  const int half = lane >> 4;          // 0 or 1 (K-half)
  const int mn   = lane & 15;          // M for A-frag, N for B-frag
  const int row0 = tile * 16;

  const float* __restrict__ arow = means + (size_t)(row0 + mn) * HH + 2 * half;
  // Clamp to a valid W row for lanes n>=2; their fragment is zeroed below.
  const float* __restrict__ wrow = W + (size_t)(mn & 1) * HH + 2 * half;
  const bool has_w = (mn < 2);
  const v2f  zf = {0.f, 0.f};

  v8f c = {};
  for (int k = 0; k < HH; k += 4) {
    const v2f a  = *(const v2f*)(arow + k);
    const v2f wv = *(const v2f*)(wrow + k);
    const v2f bf = has_w ? wv : zf;       // v_cndmask, no exec divergence
    // 8 args: (neg_a, A, neg_b, B, c_mod, C, reuse_a, reuse_b)
    c = __builtin_amdgcn_wmma_f32_16x16x4_f32(
        false, a, false, bf, (short)0, c, false, false);
  }

  if (mn < 2) {                         // only N-columns 0,1 are real logits
    const float bv = bias[mn];
#pragma unroll
    for (int r = 0; r < 8; ++r) {
      const int row = row0 + r + 8 * half;
      out[(size_t)row * 2 + mn] = c[r] + bv;
    }
  }
}

// ---------------------------------------------------------------------------
extern "C" void kernel_launch(void* const* d_in, const int* in_sizes, int n_in,
                              void* d_out, int out_size, void* d_ws,
                              size_t ws_size, hipStream_t stream) {
  (void)in_sizes; (void)n_in; (void)out_size; (void)ws_size;
  const float* hidden = (const float*)d_in[0];   // [B,S,H] fp32
  const float* W      = (const float*)d_in[1];   // [2,H]   fp32
  const float* bias   = (const float*)d_in[2];   // [2]     fp32
  const int*   ids    = (const int*)d_in[3];     // [B,S]   int32
  // d_in[4] = n_segs (== NSEG, compile-time here)

  char* ws         = (char*)d_ws;
  int*   seg_start = (int*)(ws);                 // 1024 ints
  int*   seg_end   = (int*)(ws + 4096);          // 1024 ints
  float* means     = (float*)(ws + 8192);        // 1024*1024 fp32 = 4 MB

  seg_scan_kernel<<<BB, 32, 0, stream>>>(ids, seg_start, seg_end);
  seg_mean_kernel<<<BB * NSEG, 256, 0, stream>>>(hidden, seg_start, seg_end,
                                                 means);
  wmma_classifier_kernel<<<(BB * NSEG) / 16, 32, 0, stream>>>(means, W, bias,
                                                              (float*)d_out);
}